// Encoder_33346126086886
// MI455X (gfx1250) — compile-verified
//
#include <hip/hip_runtime.h>

typedef __attribute__((ext_vector_type(2))) float v2f;
typedef __attribute__((ext_vector_type(8))) float v8f;

#define FDIM 128

// ---------------- degree / normalization ----------------

__global__ __launch_bounds__(256) void deg_init_k(float* __restrict__ deg, int n) {
  int i = blockIdx.x * 256 + threadIdx.x;
  if (i < n) deg[i] = 1.0f;  // self-loop weight
}

__global__ __launch_bounds__(256) void deg_accum_k(const int* __restrict__ dst,
                                                   const float* __restrict__ w,
                                                   float* __restrict__ deg, int e) {
  int i = blockIdx.x * 256 + threadIdx.x;
  if (i < e) atomicAdd(&deg[dst[i]], w[i]);
}

__global__ __launch_bounds__(256) void dinv_k(const float* __restrict__ deg,
                                              float* __restrict__ dinv, int n) {
  int i = blockIdx.x * 256 + threadIdx.x;
  if (i < n) {
    float d = deg[i];
    dinv[i] = d > 0.0f ? rsqrtf(d) : 0.0f;
  }
}

// ---------------- h = x @ W via V_WMMA_F32_16X16X4_F32, fused epilogue ----
// Block: 256 threads = 8 waves. W staged in LDS as K-pairs:
//   Wp[p][c'] = { W[2p][c], W[2p+1][c] },  c' = (c + 16*(p&1)) & 127
// The 16-column rotation on odd pairs makes the two half-waves (pair p for
// lanes 0..15, pair p+1 for lanes 16..31) hit disjoint halves of the 64 LDS
// banks -> conflict-free ds_load_b64, and each B fragment is one aligned
// 64-bit load (no repacking moves).
// Each wave computes a 16-row x 128-col stripe of h, then writes both
//   h[row][col]   and   emb[row][col] = b[col] + h*dinv[row]^2   (self-loop)
__global__ __launch_bounds__(256) void gcn_gemm_wmma(const float* __restrict__ x,
                                                     const float* __restrict__ W,
                                                     const float* __restrict__ bias,
                                                     const float* __restrict__ dinv,
                                                     float* __restrict__ h,
                                                     float* __restrict__ emb, int n) {
  __shared__ v2f Wp[64 * FDIM];  // 64 KB
  for (int i = threadIdx.x; i < 64 * FDIM; i += 256) {
    int p = i >> 7;
    int c = i & 127;
    v2f val;
    val.x = W[(2 * p) * FDIM + c];
    val.y = W[(2 * p + 1) * FDIM + c];
    Wp[p * FDIM + ((c + ((p & 1) << 4)) & 127)] = val;
  }
  __syncthreads();

  const int wave  = threadIdx.x >> 5;
  const int lane  = threadIdx.x & 31;
  const int mrow  = lane & 15;   // M (or N) index within tile
  const int khalf = lane >> 4;   // 0: K={0,1}, 1: K={2,3}
  const int row0  = (blockIdx.x * 8 + wave) * 16;

  // rotated column base for this lane (pair parity == khalf)
  int coff[8];
#pragma unroll
  for (int t = 0; t < 8; ++t) coff[t] = ((t * 16 + mrow + (khalf << 4)) & 127);

  v8f acc[8];
#pragma unroll
  for (int t = 0; t < 8; ++t) acc[t] = (v8f){0.f, 0.f, 0.f, 0.f, 0.f, 0.f, 0.f, 0.f};

  int arow = row0 + mrow;
  if (arow >= n) arow = n - 1;  // clamp loads; stores are guarded below
  const float* ap = x + (size_t)arow * FDIM + 2 * khalf;

#pragma unroll
  for (int k = 0; k < FDIM; k += 4) {
    const v2f a = *(const v2f*)(ap + k);     // one global_load_b64
    const int p = (k >> 1) + khalf;
#pragma unroll
    for (int t = 0; t < 8; ++t) {
      const v2f b = Wp[p * FDIM + coff[t]];  // one ds_load_b64, conflict-free
      acc[t] = __builtin_amdgcn_wmma_f32_16x16x4_f32(
          /*neg_a=*/false, a, /*neg_b=*/false, b,
          /*c_mod=*/(short)0, acc[t], /*reuse_a=*/false, /*reuse_b=*/false);
    }
  }

  // C/D layout: VGPR r -> row = row0 + r + 8*khalf, col = t*16 + (lane&15)
  const int srow = row0 + khalf * 8;
#pragma unroll
  for (int t = 0; t < 8; ++t) {
    int col = t * 16 + mrow;
    float bcol = bias[col];
#pragma unroll
    for (int r = 0; r < 8; ++r) {
      int row = srow + r;
      if (row < n) {
        float v = acc[t][r];
        float d = dinv[row];
        size_t off = (size_t)row * FDIM + col;
        h[off]   = v;
        emb[off] = bcol + v * d * d;  // self-loop term + bias
      }
    }
  }
}

// ---------------- edge scatter: one wave32 per edge ----------------
// Each lane handles 4 features (lane + 32*j): coalesced 128B gathers from h
// (L2-resident, 51MB << 192MB L2) and float atomics into emb.

__global__ __launch_bounds__(256) void scatter_k(const int* __restrict__ src,
                                                 const int* __restrict__ dst,
                                                 const float* __restrict__ w,
                                                 const float* __restrict__ dinv,
                                                 const float* __restrict__ h,
                                                 float* __restrict__ emb, int e) {
  int wid  = (int)((blockIdx.x * 256 + threadIdx.x) >> 5);
  int lane = threadIdx.x & 31;
  if (wid >= e) return;
  int s = src[wid];
  int d = dst[wid];
  float norm = dinv[s] * w[wid] * dinv[d];
  const float* hp = h + (size_t)s * FDIM;
  float* ep = emb + (size_t)d * FDIM;
#pragma unroll
  for (int j = 0; j < 4; ++j) {
    int f = lane + 32 * j;
    atomicAdd(ep + f, hp[f] * norm);
  }
}

// ---------------- relu ----------------

__global__ __launch_bounds__(256) void relu_k(const float* __restrict__ emb,
                                              float* __restrict__ out, long total) {
  long idx = (long)blockIdx.x * 256 + threadIdx.x;
  if (idx < total) out[idx] = fmaxf(emb[idx], 0.0f);
}

// ---------------- launch ----------------

extern "C" void kernel_launch(void* const* d_in, const int* in_sizes, int n_in,
                              void* d_out, int out_size, void* d_ws, size_t ws_size,
                              hipStream_t stream) {
  const float* x  = (const float*)d_in[0];
  const float* W  = (const float*)d_in[1];
  const float* b  = (const float*)d_in[2];
  // d_in[3] = level (unused scalar)
  const int*   ei = (const int*)d_in[4];
  const float* ew = (const float*)d_in[5];

  const int n = in_sizes[0] / FDIM;
  const int e = in_sizes[5];
  const int* src = ei;
  const int* dst = ei + e;

  // workspace: deg[n] | dinv[n] | h[n*128]
  float* deg  = (float*)d_ws;
  float* dinv = deg + n;
  float* h    = dinv + n;

  float* emb      = (float*)d_out;              // output 0
  float* out_relu = emb + (size_t)n * FDIM;     // output 1

  const long total = (long)n * FDIM;

  deg_init_k<<<(n + 255) / 256, 256, 0, stream>>>(deg, n);
  deg_accum_k<<<(e + 255) / 256, 256, 0, stream>>>(dst, ew, deg, e);
  dinv_k<<<(n + 255) / 256, 256, 0, stream>>>(deg, dinv, n);

  gcn_gemm_wmma<<<(n + 127) / 128, 256, 0, stream>>>(x, W, b, dinv, h, emb, n);

  scatter_k<<<(e + 7) / 8, 256, 0, stream>>>(src, dst, ew, dinv, h, emb, e);
  relu_k<<<(int)((total + 255) / 256), 256, 0, stream>>>(emb, out_relu, total);
}